// GraphMamba_59863254171816
// MI455X (gfx1250) — compile-verified
//
#include <hip/hip_runtime.h>
#include <hip/hip_bf16.h>

#define D_MODEL 256
#define D_STATE 32
#define DT_RANK 16
#define NBATCH  8
#define NNODES  1024
#define NTOK    (NBATCH * NNODES)   // 8192

typedef __attribute__((ext_vector_type(2))) float v2f;
typedef __attribute__((ext_vector_type(8))) float v8f;
typedef __attribute__((ext_vector_type(4))) unsigned int u32x4;
typedef __attribute__((ext_vector_type(8))) int i32x8;
typedef __attribute__((ext_vector_type(4))) int i32x4;

__device__ __forceinline__ v8f wmma_f32x4(v2f a, v2f b, v8f c) {
    // D = A(16x4) * B(4x16) + C(16x16), fp32 everywhere
    return __builtin_amdgcn_wmma_f32_16x16x4_f32(
        /*neg_a=*/false, a, /*neg_b=*/false, b,
        /*c_mod=*/(short)0, c, /*reuse_a=*/false, /*reuse_b=*/false);
}

__device__ __forceinline__ float silu_f(float v) {
    return v / (1.0f + __expf(-v));
}

// ---------------------------------------------------------------------------
// TDM descriptor builder + issue (ISA cdna5 §8.3/8.4).
// Loads a tile (tile_d0 contiguous elems x tile_d1 rows x tile_d2 planes) of
// 4-byte data from global into LDS, with optional LDS destination padding.
// ---------------------------------------------------------------------------
__device__ __forceinline__ void tdm_load_to_lds(
    unsigned lds_off, const void* gaddr,
    unsigned tensor_d0, unsigned tensor_d1, unsigned tensor_d2,
    unsigned tile_d0, unsigned tile_d1, unsigned tile_d2,
    unsigned long long t0_stride, unsigned long long t1_stride,
    unsigned pad_en, unsigned pad_interval, unsigned pad_amount)
{
    const unsigned long long ga = (unsigned long long)(size_t)gaddr;
    u32x4 g0;
    g0[0] = 1u;                                     // count=1, user descriptor
    g0[1] = lds_off;                                // lds_addr (bytes)
    g0[2] = (unsigned)(ga & 0xffffffffull);         // global_addr[31:0]
    g0[3] = (unsigned)((ga >> 32) & 0x01ffffffull)  // global_addr[56:32]
          | 0x80000000u;                            // type=2 ("image")
    i32x8 g1;
    g1[0] = (int)((2u << 16)                        // data_size = 4 bytes
          | (pad_en << 20) | (pad_interval << 22) | (pad_amount << 25));
    g1[1] = (int)((tensor_d0 & 0xffffu) << 16);     // [31:16]=tensor_dim0 lo
    g1[2] = (int)((tensor_d0 >> 16) | ((tensor_d1 & 0xffffu) << 16));
    g1[3] = (int)((tensor_d1 >> 16) | (tile_d0 << 16));
    g1[4] = (int)(tile_d1 | (tile_d2 << 16));
    g1[5] = (int)(t0_stride & 0xffffffffull);
    g1[6] = (int)(((t0_stride >> 32) & 0xffffull) | ((t1_stride & 0xffffull) << 16));
    g1[7] = (int)((t1_stride >> 16) & 0xffffffffull);
    i32x4 g2;
    g2[0] = (int)tensor_d2; g2[1] = 0; g2[2] = 0; g2[3] = 0;
    i32x4 g3; g3[0] = 0; g3[1] = 0; g3[2] = 0; g3[3] = 0;
#if defined(__clang_major__) && __clang_major__ >= 23
    i32x8 g4; g4[0]=0; g4[1]=0; g4[2]=0; g4[3]=0; g4[4]=0; g4[5]=0; g4[6]=0; g4[7]=0;
    __builtin_amdgcn_tensor_load_to_lds(g0, g1, g2, g3, g4, 0);
#else
    __builtin_amdgcn_tensor_load_to_lds(g0, g1, g2, g3, 0);
#endif
}

// ---------------------------------------------------------------------------
// Kernel 1: xz = (x * mask) @ Win^T    M=8192, N=512, K=256  (fp32 WMMA)
// cols [0,256) -> xc_raw, cols [256,512) -> z
// ---------------------------------------------------------------------------
__global__ __launch_bounds__(256) void k_in_proj(
    const float* __restrict__ x, const float* __restrict__ mask,
    const float* __restrict__ Win,
    float* __restrict__ xc_raw, float* __restrict__ z)
{
    const int tid  = threadIdx.x;
    const int lane = tid & 31;
    const int wave = tid >> 5;         // 0..7
    const int mw   = wave & 1;         // 2 M sub-tiles of 16
    const int nw   = wave >> 1;        // 4 N sub-tiles of 32 (2 frags)
    const int m0   = blockIdx.x * 32 + mw * 16;   // global row (b*1024+l)
    const int n0   = blockIdx.y * 128 + nw * 32;  // global col in [0,512)
    const int lm   = lane & 15;
    const int lk   = (lane >> 4) * 2;  // K offset 0 or 2

    v8f acc0 = {0,0,0,0,0,0,0,0};
    v8f acc1 = {0,0,0,0,0,0,0,0};

    const float maskv = mask[m0 + lm];
    const float* __restrict__ arow = x   + (size_t)(m0 + lm) * D_MODEL;
    const float* __restrict__ b0   = Win + (size_t)(n0 + lm)      * D_MODEL;
    const float* __restrict__ b1   = Win + (size_t)(n0 + 16 + lm) * D_MODEL;

    for (int k = 0; k < D_MODEL; k += 4) {
        v2f a;  a.x  = arow[k + lk] * maskv; a.y  = arow[k + lk + 1] * maskv;
        v2f bb0; bb0.x = b0[k + lk];         bb0.y = b0[k + lk + 1];
        v2f bb1; bb1.x = b1[k + lk];         bb1.y = b1[k + lk + 1];
        acc0 = wmma_f32x4(a, bb0, acc0);
        acc1 = wmma_f32x4(a, bb1, acc1);
    }

    const int rbase = m0 + (lane >> 4) * 8;
    const int cA = n0 + lm;
    const int cB = n0 + 16 + lm;
    #pragma unroll
    for (int v = 0; v < 8; ++v) {
        const size_t r = (size_t)(rbase + v);
        if (cA < D_MODEL) xc_raw[r * D_MODEL + cA] = acc0[v];
        else              z[r * D_MODEL + (cA - D_MODEL)] = acc0[v];
        if (cB < D_MODEL) xc_raw[r * D_MODEL + cB] = acc1[v];
        else              z[r * D_MODEL + (cB - D_MODEL)] = acc1[v];
    }
}

// ---------------------------------------------------------------------------
// Kernel 2: depthwise causal conv (k=4, left pad 3) + bias + SiLU
// ---------------------------------------------------------------------------
__global__ __launch_bounds__(256) void k_conv(
    const float* __restrict__ xc_raw, const float* __restrict__ conv_w,
    const float* __restrict__ conv_b, float* __restrict__ xc_conv)
{
    const int idx = blockIdx.x * 256 + threadIdx.x;   // b*L*D + l*D + d
    const int d   = idx & (D_MODEL - 1);
    const int l   = (idx >> 8) & (NNODES - 1);
    const int b   = idx >> 18;
    float s = conv_b[d];
    #pragma unroll
    for (int j = 0; j < 4; ++j) {
        const int ll = l - 3 + j;
        if (ll >= 0)
            s += xc_raw[(((size_t)b << 10) + ll) * D_MODEL + d] * conv_w[d * 4 + j];
    }
    xc_conv[idx] = silu_f(s);
}

// ---------------------------------------------------------------------------
// Kernel 3: x_dbl = xc @ Wxp^T (80 outs) ; dt = softplus(dt_in @ Wdt^T + bdt)
// one workgroup per token row
// ---------------------------------------------------------------------------
__global__ __launch_bounds__(256) void k_xproj(
    const float* __restrict__ xc_conv, const float* __restrict__ Wxp,
    const float* __restrict__ Wdt, const float* __restrict__ bdt,
    float* __restrict__ x_dbl, float* __restrict__ dt)
{
    __shared__ float xrow[D_MODEL];
    __shared__ float dtin[DT_RANK];
    const int bl  = blockIdx.x;          // 0..8191
    const int tid = threadIdx.x;

    xrow[tid] = xc_conv[(size_t)bl * D_MODEL + tid];
    __syncthreads();

    if (tid < 80) {
        const float* __restrict__ w = Wxp + (size_t)tid * D_MODEL;
        float s = 0.0f;
        for (int k = 0; k < D_MODEL; ++k) s += xrow[k] * w[k];
        x_dbl[(size_t)bl * 80 + tid] = s;
        if (tid < DT_RANK) dtin[tid] = s;
    }
    __syncthreads();

    float s = bdt[tid];
    const float* __restrict__ w = Wdt + (size_t)tid * DT_RANK;
    #pragma unroll
    for (int r = 0; r < DT_RANK; ++r) s += dtin[r] * w[r];
    const float sp = (s > 20.0f) ? s : __logf(1.0f + __expf(s));
    dt[(size_t)bl * D_MODEL + tid] = sp;
}

// ---------------------------------------------------------------------------
// Kernel 4: selective scan. wave32 == D_STATE: one wave owns one (b,d),
// lane == state index s. Fused epilogue: y = (scan + xc*D) * silu(z).
// ---------------------------------------------------------------------------
__global__ __launch_bounds__(256) void k_scan(
    const float* __restrict__ dt, const float* __restrict__ xc_conv,
    const float* __restrict__ x_dbl, const float* __restrict__ z,
    const float* __restrict__ A_log, const float* __restrict__ D_skip,
    float* __restrict__ y)
{
    const int tid  = threadIdx.x;
    const int lane = tid & 31;                 // s
    const int wave = tid >> 5;                 // 0..7
    const int b    = blockIdx.y;               // 0..7
    const int d    = blockIdx.x * 8 + wave;    // 0..255

    const float Arow = -__expf(A_log[d * D_STATE + lane]);
    const float Dv   = D_skip[d];

    const float* __restrict__ dtp = dt      + (size_t)b * NNODES * D_MODEL + d;
    const float* __restrict__ xcp = xc_conv + (size_t)b * NNODES * D_MODEL + d;
    const float* __restrict__ zp  = z       + (size_t)b * NNODES * D_MODEL + d;
    const float* __restrict__ xd  = x_dbl   + (size_t)b * NNODES * 80;
    float* __restrict__ yp        = y       + (size_t)b * NNODES * D_MODEL + d;

    float h = 0.0f;
    for (int l = 0; l < NNODES; ++l) {
        const float dtv = dtp[(size_t)l * D_MODEL];
        const float xcv = xcp[(size_t)l * D_MODEL];
        const float Bv  = xd[l * 80 + DT_RANK + lane];
        const float Cv  = xd[l * 80 + DT_RANK + D_STATE + lane];
        h = __expf(dtv * Arow) * h + (dtv * xcv) * Bv;
        float yv = h * Cv;
        #pragma unroll
        for (int off = 16; off > 0; off >>= 1)
            yv += __shfl_down(yv, off, 32);
        if (lane == 0) {
            const float zv = zp[(size_t)l * D_MODEL];
            yp[(size_t)l * D_MODEL] = (yv + xcv * Dv) * silu_f(zv);
        }
    }
}

// ---------------------------------------------------------------------------
// Kernel 5: attn = mask * (y @ Wout^T)   M=8192, N=256, K=256 (fp32 WMMA)
// ---------------------------------------------------------------------------
__global__ __launch_bounds__(256) void k_out_proj(
    const float* __restrict__ y, const float* __restrict__ mask,
    const float* __restrict__ Wout, float* __restrict__ attn)
{
    const int tid  = threadIdx.x;
    const int lane = tid & 31;
    const int wave = tid >> 5;
    const int mw   = wave & 1;
    const int nw   = wave >> 1;
    const int m0   = blockIdx.x * 32 + mw * 16;
    const int n0   = blockIdx.y * 128 + nw * 32;
    const int lm   = lane & 15;
    const int lk   = (lane >> 4) * 2;

    v8f acc0 = {0,0,0,0,0,0,0,0};
    v8f acc1 = {0,0,0,0,0,0,0,0};

    const float* __restrict__ arow = y    + (size_t)(m0 + lm) * D_MODEL;
    const float* __restrict__ b0   = Wout + (size_t)(n0 + lm)      * D_MODEL;
    const float* __restrict__ b1   = Wout + (size_t)(n0 + 16 + lm) * D_MODEL;

    for (int k = 0; k < D_MODEL; k += 4) {
        v2f a;  a.x  = arow[k + lk]; a.y  = arow[k + lk + 1];
        v2f bb0; bb0.x = b0[k + lk]; bb0.y = b0[k + lk + 1];
        v2f bb1; bb1.x = b1[k + lk]; bb1.y = b1[k + lk + 1];
        acc0 = wmma_f32x4(a, bb0, acc0);
        acc1 = wmma_f32x4(a, bb1, acc1);
    }

    const int rbase = m0 + (lane >> 4) * 8;
    #pragma unroll
    for (int v = 0; v < 8; ++v) {
        const int r = rbase + v;
        const float mv = mask[r];
        attn[(size_t)r * D_MODEL + n0 + lm]      = acc0[v] * mv;
        attn[(size_t)r * D_MODEL + n0 + 16 + lm] = acc1[v] * mv;
    }
}

// ---------------------------------------------------------------------------
// Kernel 6: out = attn + (Σ_t wE[t]*adj[t]) @ attn   (per batch)
// M=1024, N=256, K=1024 per b. 32x256 output tile per WG, K chunked by 16.
// TDM (tensor_load_to_lds) double-buffered staging:
//  - attn tile (16x256 f32) via a 2D descriptor with LDS pad (row 256->272)
//  - all 3 edge-type adj tiles (3 x 32x16 f32) via one 3D descriptor whose
//    tensor_dim1_stride is the edge-type stride
// wave 0 issues DMA / s_wait_tensorcnt; all waves sync via barrier; WMMA
// compute on buffer c overlaps TDM for buffer c+1.
// ---------------------------------------------------------------------------
#define KC 16

__global__ __launch_bounds__(256) void k_edge(
    const float* __restrict__ adj, const float* __restrict__ wE,
    const float* __restrict__ attn, float* __restrict__ out)
{
    __shared__ float S_l[2][KC][272];      // attn tile  [j][d], padded rows
    __shared__ float Araw[2][3][32][KC];   // raw adj tiles per edge type
    __shared__ float A_l[2][32][18];       // combined Aeff tile, padded

    const int b    = blockIdx.y;
    const int i0   = blockIdx.x * 32;
    const int tid  = threadIdx.x;
    const int lane = tid & 31;
    const int wave = tid >> 5;
    const int mw   = wave & 1;       // 2 M sub-tiles of 16
    const int nw   = wave >> 1;      // 4 N sub-tiles of 64 (4 frags each)
    const int lm   = lane & 15;
    const int lk   = (lane >> 4) * 2;

    const float w0 = wE[0], w1 = wE[1], w2 = wE[2];
    const unsigned long long tstr = (unsigned long long)NBATCH * NNODES * NNODES;
    const float* __restrict__ adj0 = adj  + (size_t)b * NNODES * NNODES;
    const float* __restrict__ Sb   = attn + (size_t)b * NNODES * D_MODEL;

    v8f acc[4];
    #pragma unroll
    for (int f = 0; f < 4; ++f) acc[f] = (v8f){0,0,0,0,0,0,0,0};

    auto issue_chunk = [&](int buf, int j0) {
        // attn tile: 16 rows (j) x 256 cols (d); pad LDS rows 256 -> 272 DW
        tdm_load_to_lds((unsigned)(size_t)&S_l[buf][0][0],
                        Sb + (size_t)j0 * D_MODEL,
                        /*tensor*/ 256u, 1024u, 0u,
                        /*tile  */ 256u, KC, 0u,
                        /*stride*/ 256ull, 0ull,
                        /*pad   */ 1u, 7u, 15u);   // every 256 DW, +16 DW
        // adj tiles: x=16 (j), y=32 (i), z=3 (edge type)
        tdm_load_to_lds((unsigned)(size_t)&Araw[buf][0][0][0],
                        adj0 + (size_t)i0 * NNODES + j0,
                        /*tensor*/ 1024u, 1024u, 3u,
                        /*tile  */ KC, 32u, 3u,
                        /*stride*/ 1024ull, tstr,
                        /*pad   */ 0u, 0u, 0u);
    };

    if (wave == 0) issue_chunk(0, 0);

    const int nchunks = NNODES / KC;   // 64
    for (int c = 0; c < nchunks; ++c) {
        const int buf = c & 1;
        if (wave == 0) __builtin_amdgcn_s_wait_tensorcnt((short)0);
        __syncthreads();   // data for chunk c resident, all waves past compute c-1

        // combine 3 edge types into Aeff (32 x KC), padded rows
        {
            const int j = tid & 15;
            const int i = tid >> 4;    // 0..15
            #pragma unroll
            for (int e = 0; e < 2; ++e) {
                const int ii = i + e * 16;
                A_l[buf][ii][j] = w0 * Araw[buf][0][ii][j]
                                + w1 * Araw[buf][1][ii][j]
                                + w2 * Araw[buf][2][ii][j];
            }
        }
        __syncthreads();

        // overlap: DMA next chunk into the other buffer while we WMMA
        if (wave == 0 && c + 1 < nchunks) issue_chunk(buf ^ 1, (c + 1) * KC);

        #pragma unroll
        for (int kk = 0; kk < KC; kk += 4) {
            v2f a;
            a.x = A_l[buf][mw * 16 + lm][kk + lk];
            a.y = A_l[buf][mw * 16 + lm][kk + lk + 1];
            #pragma unroll
            for (int f = 0; f < 4; ++f) {
                const int n = nw * 64 + f * 16 + lm;
                v2f bb;
                bb.x = S_l[buf][kk + lk][n];
                bb.y = S_l[buf][kk + lk + 1][n];
                acc[f] = wmma_f32x4(a, bb, acc[f]);
            }
        }
    }

    // ---- epilogue: out = attn + B_E ----
    const int rbase = i0 + mw * 16 + (lane >> 4) * 8;
    #pragma unroll
    for (int f = 0; f < 4; ++f) {
        const int n = nw * 64 + f * 16 + lm;
        #pragma unroll
        for (int v = 0; v < 8; ++v) {
            const int r = rbase + v;
            const size_t o = ((size_t)b * NNODES + r) * D_MODEL + n;
            out[o] = attn[o] + acc[f][v];
        }
    }
}

// ---------------------------------------------------------------------------
extern "C" void kernel_launch(void* const* d_in, const int* in_sizes, int n_in,
                              void* d_out, int out_size, void* d_ws, size_t ws_size,
                              hipStream_t stream) {
    (void)in_sizes; (void)n_in; (void)out_size; (void)ws_size;
    const float* x      = (const float*)d_in[0];
    const float* mask   = (const float*)d_in[1];
    const float* adj    = (const float*)d_in[2];
    const float* wE     = (const float*)d_in[3];
    const float* Win    = (const float*)d_in[4];
    const float* conv_w = (const float*)d_in[5];
    const float* conv_b = (const float*)d_in[6];
    const float* Wxp    = (const float*)d_in[7];
    const float* Wdt    = (const float*)d_in[8];
    const float* bdt    = (const float*)d_in[9];
    const float* A_log  = (const float*)d_in[10];
    const float* D_skip = (const float*)d_in[11];
    const float* Wout   = (const float*)d_in[12];
    float* out = (float*)d_out;

    float* ws   = (float*)d_ws;
    float* buf0 = ws;                          // xc_raw -> y (reused)
    float* buf1 = buf0 + (size_t)NTOK * 256;   // z -> attn (reused)
    float* buf2 = buf1 + (size_t)NTOK * 256;   // xc_conv
    float* buf3 = buf2 + (size_t)NTOK * 256;   // dt
    float* buf4 = buf3 + (size_t)NTOK * 256;   // x_dbl (80 per token)

    k_in_proj <<<dim3(NTOK / 32, 4), 256, 0, stream>>>(x, mask, Win, buf0, buf1);
    k_conv    <<<NTOK, 256, 0, stream>>>(buf0, conv_w, conv_b, buf2);
    k_xproj   <<<NTOK, 256, 0, stream>>>(buf2, Wxp, Wdt, bdt, buf4, buf3);
    k_scan    <<<dim3(32, NBATCH), 256, 0, stream>>>(buf3, buf2, buf4, buf1,
                                                     A_log, D_skip, buf0);
    k_out_proj<<<dim3(NTOK / 32, 2), 256, 0, stream>>>(buf0, mask, Wout, buf1);
    k_edge    <<<dim3(NNODES / 32, NBATCH), 256, 0, stream>>>(adj, wE, buf1, out);
}